// KMaxPooling1D_85822036509201
// MI455X (gfx1250) — compile-verified
//
#include <hip/hip_runtime.h>
#include <stdint.h>

#define HAS_TDM __has_builtin(__builtin_amdgcn_tensor_load_to_lds)
#define HAS_WTC __has_builtin(__builtin_amdgcn_s_wait_tensorcnt)

typedef unsigned int v4u __attribute__((ext_vector_type(4)));
typedef int          v8i __attribute__((ext_vector_type(8)));
typedef int          v4i __attribute__((ext_vector_type(4)));

static constexpr int Bc    = 32;    // batch
static constexpr int Sc    = 8192;  // sequence
static constexpr int Cc    = 256;   // channels
static constexpr int CB    = 32;    // channels per block (= lanes per wave)
static constexpr int NW    = 8;     // waves per block (256 threads)
static constexpr int TROWS = 256;   // s-rows per TDM tile (32 KB tile)
static constexpr int NT    = Sc / TROWS;

struct SMem {
  float    tile[2][TROWS * CB];   // 64 KB double-buffered TDM landing zone
  unsigned hist[256 * CB];        // 32 KB per-column radix histograms (bin-major)
  unsigned prefix[CB];
  unsigned kneed[CB];
  unsigned tkey[CB];
  int      skipE[CB];
  unsigned cntG[NW][CB];
  unsigned cntE[NW][CB];
  unsigned carG[CB];
  unsigned carE[CB];
};

// Order-preserving float -> uint key (ascending float == ascending key)
__device__ __forceinline__ unsigned mono(float f) {
  unsigned u = __float_as_uint(f);
  return (u & 0x80000000u) ? ~u : (u | 0x80000000u);
}

__device__ __forceinline__ void wait_tensor0() {
#if HAS_WTC
  __builtin_amdgcn_s_wait_tensorcnt(0);
#else
  asm volatile("s_wait_tensorcnt 0x0" ::: "memory");
#endif
}
__device__ __forceinline__ void wait_tensor1() {
#if HAS_WTC
  __builtin_amdgcn_s_wait_tensorcnt(1);
#else
  asm volatile("s_wait_tensorcnt 0x1" ::: "memory");
#endif
}

#if HAS_TDM
// Issue one TDM 2D tile load: 32 fp32 (dim0) x TROWS rows (dim1), row stride 256
// elements, packed densely into LDS at lds_off. D# layout per CDNA5 ISA §8.
// This toolchain exposes the 6-arg builtin:
//   (uint32x4 g0, int32x8 g1, int32x4 g2, int32x4 g3, int32x8 g4, i32 cpol)
__device__ __forceinline__ void tdm_issue(uint64_t gaddr, unsigned lds_off) {
  v4u g0;
  g0.x = 1u;                                     // count=1, user descriptor
  g0.y = lds_off;                                // LDS byte address
  g0.z = (unsigned)(gaddr & 0xFFFFFFFFull);      // global_addr[31:0]
  g0.w = ((unsigned)(gaddr >> 32) & 0x01FFFFFFu) // global_addr[56:32]
         | (2u << 30);                           // type=2 ("image")
  v8i g1;
  g1[0] = (int)(2u << 16);                       // data_size=4B; no pad/iter/multicast
  g1[1] = 0;                                     // abar=0; tensor_dim0(=1<<20) low16
  g1[2] = 0x0010;                                // tensor_dim0 hi16; tensor_dim1(=1<<20) low16
  g1[3] = (int)(0x0010u | ((unsigned)CB << 16)); // tensor_dim1 hi16; tile_dim0=32
  g1[4] = TROWS;                                 // tile_dim1=256; tile_dim2=0
  g1[5] = Cc;                                    // tensor_dim0_stride=256 elems (low32)
  g1[6] = 0;                                     // stride0 hi16; stride1 low16
  g1[7] = 0;
  v4i z4 = {0, 0, 0, 0};
  v8i z8 = {0, 0, 0, 0, 0, 0, 0, 0};
  __builtin_amdgcn_tensor_load_to_lds(g0, g1, z4, z4, z8, 0);
}
#endif

// Stream all NT tiles of this block's [Sc x 32-channel] stripe through LDS,
// double-buffered via TDM, calling f(tile_index, tile_ptr) on each.
template <typename F>
__device__ __forceinline__ void for_tiles(SMem& sm, const float* gbase, F&& f) {
  const int tid = threadIdx.x;
#if HAS_TDM
  const bool issuer = (tid < 32);  // wave 0 drives the TDM pipeline
  const unsigned off0 = (unsigned)((uintptr_t)&sm.tile[0][0] - (uintptr_t)&sm);
  const unsigned off1 = (unsigned)((uintptr_t)&sm.tile[1][0] - (uintptr_t)&sm);
  if (issuer) tdm_issue((uint64_t)(uintptr_t)gbase, off0);
  for (int t = 0; t < NT; ++t) {
    if (issuer) {
      if (t + 1 < NT) {
        tdm_issue((uint64_t)(uintptr_t)(gbase + (size_t)(t + 1) * TROWS * Cc),
                  ((t + 1) & 1) ? off1 : off0);
        wait_tensor1();   // in-order per wave: tile t is complete
      } else {
        wait_tensor0();
      }
    }
    __syncthreads();      // tile t visible to all waves
    f(t, sm.tile[t & 1]);
    __syncthreads();      // everyone done before buffer is overwritten
  }
#else
  for (int t = 0; t < NT; ++t) {
    __syncthreads();
    const float* g = gbase + (size_t)t * TROWS * Cc;
    for (int i = tid; i < TROWS * CB; i += NW * 32)
      sm.tile[0][i] = g[(size_t)(i >> 5) * Cc + (i & 31)];
    __syncthreads();
    f(t, sm.tile[0]);
    __syncthreads();
  }
#endif
}

__global__ __launch_bounds__(NW * 32) void kmax_kernel(
    const float* __restrict__ x, const int* __restrict__ kp,
    float* __restrict__ out) {
  __shared__ SMem sm;
  const int tid  = threadIdx.x;
  const int lane = tid & 31;          // channel within block
  const int wave = tid >> 5;          // s-segment within tile
  const int b    = blockIdx.x >> 3;
  const int c0   = (blockIdx.x & 7) * CB;
  const int k    = *kp;

  const float* gbase = x + (size_t)b * Sc * Cc + c0;

  // ---- Phase 1: radix select (MSB->LSB byte) threshold per column ----
  if (tid < CB) { sm.prefix[tid] = 0u; sm.kneed[tid] = (unsigned)k; }
  for (int d = 3; d >= 0; --d) {
    for (int i = tid; i < 256 * CB; i += NW * 32) sm.hist[i] = 0u;
    __syncthreads();
    const unsigned pfx = sm.prefix[lane];
    const int hsh = 8 * (d + 1);
    for_tiles(sm, gbase, [&](int, const float* buf) {
      const int r0 = wave * (TROWS / NW);
#pragma unroll 4
      for (int r = r0; r < r0 + TROWS / NW; ++r) {
        const unsigned key = mono(buf[r * CB + lane]);
        const bool ok = (d == 3) || ((key >> hsh) == pfx);
        if (ok) atomicAdd(&sm.hist[((key >> (8 * d)) & 0xFFu) * CB + lane], 1u);
      }
    });
    if (tid < CB) {
      const unsigned need = sm.kneed[tid];
      unsigned cum = 0, cnt = 0;
      int chosen = 0;
      for (int bin = 255; bin >= 0; --bin) {
        cnt = sm.hist[bin * CB + tid];
        if (cum + cnt >= need) { chosen = bin; break; }
        cum += cnt;
      }
      const unsigned pfx2 = (sm.prefix[tid] << 8) | (unsigned)chosen;
      sm.prefix[tid] = pfx2;
      const unsigned m = need - cum;   // still needed inside chosen bin
      sm.kneed[tid] = m;
      if (d == 0) {
        sm.tkey[tid]  = pfx2;                 // exact threshold key
        sm.skipE[tid] = (int)cnt - (int)m;    // #earliest ties to drop (stable-sort rule)
      }
    }
    __syncthreads();
  }

  // ---- Phase 2: ordered collect with exact output positions ----
  const unsigned tk  = sm.tkey[lane];
  const int      skE = sm.skipE[lane];
  if (tid < CB) { sm.carG[tid] = 0u; sm.carE[tid] = 0u; }
  float* obase = out + (size_t)b * k * Cc + (c0 + lane);

  for_tiles(sm, gbase, [&](int, const float* buf) {
    const int r0 = wave * (TROWS / NW);
    unsigned gc = 0, ec = 0;
    for (int r = r0; r < r0 + TROWS / NW; ++r) {
      const unsigned key = mono(buf[r * CB + lane]);
      gc += (key > tk);
      ec += (key == tk);
    }
    sm.cntG[wave][lane] = gc;
    sm.cntE[wave][lane] = ec;
    __syncthreads();
    unsigned gs = sm.carG[lane], es = sm.carE[lane];
    for (int w = 0; w < NW; ++w)
      if (w < wave) { gs += sm.cntG[w][lane]; es += sm.cntE[w][lane]; }
    __syncthreads();
    if (wave == NW - 1) { sm.carG[lane] = gs + gc; sm.carE[lane] = es + ec; }
    unsigned g = gs, e = es;
    for (int r = r0; r < r0 + TROWS / NW; ++r) {
      const float v = buf[r * CB + lane];
      const unsigned key = mono(v);
      if (key > tk) {
        int extra = (int)e - skE; if (extra < 0) extra = 0;
        obase[(size_t)(g + (unsigned)extra) * Cc] = v;
        ++g;
      } else if (key == tk) {
        if ((int)e >= skE)
          obase[(size_t)(g + (unsigned)((int)e - skE)) * Cc] = v;
        ++e;
      }
    }
  });
}

extern "C" void kernel_launch(void* const* d_in, const int* in_sizes, int n_in,
                              void* d_out, int out_size, void* d_ws, size_t ws_size,
                              hipStream_t stream) {
  (void)in_sizes; (void)n_in; (void)out_size; (void)d_ws; (void)ws_size;
  const float* x  = (const float*)d_in[0];
  const int*   kp = (const int*)d_in[1];
  float*       out = (float*)d_out;
  dim3 grid(Bc * (Cc / CB));   // 32 batches x 8 channel-blocks = 256 WGs
  kmax_kernel<<<grid, NW * 32, 0, stream>>>(x, kp, out);
}